// GaussianKernel_29884382445609
// MI455X (gfx1250) — compile-verified
//
#include <hip/hip_runtime.h>

// KDE via WMMA f32 16x16x4 on gfx1250 (CDNA5, wave32).
// log p(x_i) = logsumexp_j( -0.5 * ||x_i - y_j||^2 * w_j^2 - Z ),  h = 1
// dist2 via GEMM expansion: ||x||^2 - 2 x.y + ||y||^2, cross term on the
// f32 matrix pipe (V_WMMA_F32_16X16X4_F32) to keep exact f32 numerics.

typedef __attribute__((ext_vector_type(2))) float v2f;
typedef __attribute__((ext_vector_type(8))) float v8f;

#define NT   2048   // test rows
#define NTR  4096   // train rows
#define DIM  64     // features
#define NCT  (NTR / 16)   // 256 column tiles
#define WAVES 8

__global__ __launch_bounds__(256) void kde_wmma_kernel(
    const float* __restrict__ test,
    const float* __restrict__ train,
    const float* __restrict__ weights,
    float* __restrict__ out) {
  // Z = 0.5*d*log(2*pi) + d*log(h) + log(n),  h=1, d=64, n=4096
  const float Z = 32.0f * 1.8378770664093453f + 8.317766166719343f;

  const int lane = threadIdx.x & 31;
  const int wave = threadIdx.x >> 5;
  const int half = lane >> 4;     // selects K-pair {0,1} vs {2,3} per WMMA step
  const int l16  = lane & 15;     // M for A, N for B, N for C/D
  const int row_base = blockIdx.x * 16;

  // ---- A tile: 16 rows x 64 feats, resident in 16 v2f regs per lane ----
  // ISA 32-bit A 16x4 layout: lanes 0-15 hold K={0,1}, lanes 16-31 K={2,3};
  // row M = lane%16. For k_base = 4*kb, lane loads feats 4*kb + 2*half + {0,1}.
  v2f a[16];
  const float* arow = test + (size_t)(row_base + l16) * DIM + 2 * half;
#pragma unroll
  for (int kb = 0; kb < 16; ++kb)
    a[kb] = *(const v2f*)(arow + 4 * kb);

  // Row squared-norms from A regs: lane holds half of row l16's features.
  float asq = 0.f;
#pragma unroll
  for (int kb = 0; kb < 16; ++kb)
    asq += a[kb].x * a[kb].x + a[kb].y * a[kb].y;
  float rowsq = asq + __shfl_xor(asq, 16, 32);        // full ||x_{l16}||^2

  // C/D layout: VGPR r holds row M=r (lanes 0-15) / M=r+8 (lanes 16-31).
  // Redistribute row norms into that mapping.
  float tq[8];
#pragma unroll
  for (int r = 0; r < 8; ++r)
    tq[r] = __shfl(rowsq, r + 8 * half, 32);

  // Streaming logsumexp state per (row-slot, lane-column)
  float m_acc[8], s_acc[8];
#pragma unroll
  for (int r = 0; r < 8; ++r) { m_acc[r] = -__builtin_inff(); s_acc[r] = 0.f; }

  for (int ct = wave; ct < NCT; ct += WAVES) {
    const int col = ct * 16 + l16;

    // ---- B tile: 4x16 per WMMA step; N = lane%16, K-pair by lane half ----
    v2f b[16];
    const float* brow = train + (size_t)col * DIM + 2 * half;
#pragma unroll
    for (int kb = 0; kb < 16; ++kb)
      b[kb] = *(const v2f*)(brow + 4 * kb);

    // Prefetch next tile's B data for this wave (global_prefetch_b8).
    if (ct + WAVES < NCT)
      __builtin_prefetch(train + (size_t)(col + WAVES * 16) * DIM + 2 * half, 0, 1);

    // ---- cross = test_tile * train_tile^T via 16 x V_WMMA_F32_16X16X4_F32 ----
    v8f c = {};
#pragma unroll
    for (int kb = 0; kb < 16; ++kb)
      c = __builtin_amdgcn_wmma_f32_16x16x4_f32(
          /*neg_a=*/false, a[kb], /*neg_b=*/false, b[kb],
          /*c_mod=*/(short)0, c, /*reuse_a=*/false, /*reuse_b=*/false);

    // Column norm from B regs (half per lane-half) + per-column scale.
    float bsq = 0.f;
#pragma unroll
    for (int kb = 0; kb < 16; ++kb)
      bsq += b[kb].x * b[kb].x + b[kb].y * b[kb].y;
    const float cq = bsq + __shfl_xor(bsq, 16, 32);   // full ||y_col||^2
    float sc = weights[col];
    sc = sc * sc;                                     // (w/h)^2, h=1

    // Epilogue: dist2 -> log-kernel value -> online logsumexp update.
#pragma unroll
    for (int r = 0; r < 8; ++r) {
      float d2 = tq[r] - 2.0f * c[r] + cq;
      d2 = fmaxf(d2, 0.0f);
      const float x  = fmaf(-0.5f * sc, d2, -Z);
      const float nm = fmaxf(m_acc[r], x);
      s_acc[r] = s_acc[r] * __expf(m_acc[r] - nm) + __expf(x - nm);
      m_acc[r] = nm;
    }
  }

  // ---- reduce logsumexp state across the 16 N-lanes (within each half) ----
#pragma unroll
  for (int off = 1; off < 16; off <<= 1) {
#pragma unroll
    for (int r = 0; r < 8; ++r) {
      const float om = __shfl_xor(m_acc[r], off, 32);
      const float os = __shfl_xor(s_acc[r], off, 32);
      const float nm = fmaxf(m_acc[r], om);
      s_acc[r] = s_acc[r] * __expf(m_acc[r] - nm) + os * __expf(om - nm);
      m_acc[r] = nm;
    }
  }

  // ---- combine the 8 wave-partials via LDS ----
  __shared__ float red_m[WAVES][16];
  __shared__ float red_s[WAVES][16];
  if (l16 == 0) {
#pragma unroll
    for (int r = 0; r < 8; ++r) {
      red_m[wave][r + 8 * half] = m_acc[r];
      red_s[wave][r + 8 * half] = s_acc[r];
    }
  }
  __syncthreads();

  if (threadIdx.x < 16) {
    float m = red_m[0][threadIdx.x];
    float s = red_s[0][threadIdx.x];
#pragma unroll
    for (int wv = 1; wv < WAVES; ++wv) {
      const float om = red_m[wv][threadIdx.x];
      const float os = red_s[wv][threadIdx.x];
      const float nm = fmaxf(m, om);
      s = s * __expf(m - nm) + os * __expf(om - nm);
      m = nm;
    }
    out[row_base + threadIdx.x] = m + logf(s);
  }
}

extern "C" void kernel_launch(void* const* d_in, const int* in_sizes, int n_in,
                              void* d_out, int out_size, void* d_ws, size_t ws_size,
                              hipStream_t stream) {
  (void)in_sizes; (void)n_in; (void)d_ws; (void)ws_size; (void)out_size;
  const float* test    = (const float*)d_in[0];   // [2048, 64] f32
  const float* train   = (const float*)d_in[1];   // [4096, 64] f32
  const float* weights = (const float*)d_in[2];   // [4096]     f32
  float* out = (float*)d_out;                     // [2048]     f32

  kde_wmma_kernel<<<NT / 16, 256, 0, stream>>>(test, train, weights, out);
}